// R_HGNN_Layer_8959301780057
// MI455X (gfx1250) — compile-verified
//
#include <hip/hip_runtime.h>

// ---------------- problem constants (match reference) ----------------
constexpr int R_REL   = 3;
constexpr int N_NODES = 100000;   // 6250 * 16 exactly
constexpr int E_EDGES = 400000;
constexpr int IN_DIM  = 128;
constexpr int H       = 8;
constexpr int D       = 16;
constexpr int HD      = H * D;    // 128
constexpr int REL_IN  = 64;
constexpr int REL_HID = 8;
constexpr float NEG   = 0.2f;

constexpr int TILES   = 10;       // node tiles (16 rows) per block; 6250 = 625*10
constexpr int NTILE   = N_NODES / 16;      // 6250
constexpr int NBLK    = NTILE / TILES;     // 625

typedef __attribute__((ext_vector_type(2))) float v2f;
typedef __attribute__((ext_vector_type(8))) float v8f;

// ---------------- atomic helpers (native CDNA5 paths) ----------------
__device__ __forceinline__ void atomicAddF(float* p, float v) {
    __hip_atomic_fetch_add(p, v, __ATOMIC_RELAXED, __HIP_MEMORY_SCOPE_AGENT);
}

__device__ __forceinline__ void atomicMaxF(float* addr, float val) {
    // ordered-int trick; storage initialized to -inf (0xFF800000)
    if (val >= 0.0f) {
        atomicMax((int*)addr, __float_as_int(val));
    } else {
        atomicMin((unsigned int*)addr, (unsigned int)__float_as_int(val));
    }
}

__device__ __forceinline__ float leaky(float v) { return v > 0.0f ? v : NEG * v; }

// ---- async global->LDS tile streaming (GLOBAL_LOAD_ASYNC_TO_LDS_B128, GVS) ----
// Each call moves 32 bytes for this lane: two B128 ops; INST_OFFSET is applied to
// both the LDS and global addresses (ISA 15.18.3), so one (lds,voff) pair serves both.
__device__ __forceinline__ void async_load_32B(uint32_t lds_addr,
                                               const float* __restrict__ gbase,
                                               uint32_t gbyte) {
    asm volatile(
        "global_load_async_to_lds_b128 %0, %1, %2\n\t"
        "global_load_async_to_lds_b128 %0, %1, %2 offset:16"
        :
        : "v"(lds_addr), "v"(gbyte), "s"(gbase)
        : "memory");
}
#define WAIT_ASYNCCNT(imm) asm volatile("s_wait_asynccnt " #imm ::: "memory")

__device__ __forceinline__ uint32_t lds_off32(const void* p) {
    // low 32 bits of a flat LDS-aperture address == group-segment byte offset
    return (uint32_t)(uintptr_t)p;
}

// ---------------- tiny kernel: rw = emb @ rel_attn_w, rel_out = emb @ Wp + bp
__global__ __launch_bounds__(256) void k_small(
    const float* __restrict__ rel_emb,     // (R,64)
    const float* __restrict__ rel_attn_w,  // (R,64,256)
    const float* __restrict__ Wp,          // (R,64,64)
    const float* __restrict__ bp,          // (R,64)
    float* __restrict__ rwbuf,             // (R,256)
    float* __restrict__ rel_out)           // (R,64)
{
    int tid = threadIdx.x;
    for (int i = tid; i < R_REL * 2 * HD; i += 256) {   // 768
        int r = i >> 8, k = i & 255;
        float s = 0.0f;
        for (int j = 0; j < REL_IN; ++j)
            s += rel_emb[r * REL_IN + j] * rel_attn_w[(r * REL_IN + j) * (2 * HD) + k];
        rwbuf[i] = s;
    }
    for (int i = tid; i < R_REL * H * REL_HID; i += 256) {  // 192
        int r = i / (H * REL_HID), o = i % (H * REL_HID);
        float s = bp[i];
        for (int j = 0; j < REL_IN; ++j)
            s += rel_emb[r * REL_IN + j] * Wp[(r * REL_IN + j) * (H * REL_HID) + o];
        rel_out[i] = s;
    }
}

// ---------------- init: segmax=-inf, segsum=0, agg(out region)=0 ----------------
__global__ __launch_bounds__(256) void k_init(
    float* __restrict__ segmax, float* __restrict__ segsum, float* __restrict__ agg)
{
    size_t i = (size_t)blockIdx.x * blockDim.x + threadIdx.x;
    const size_t nh  = (size_t)R_REL * N_NODES * H;
    const size_t nhd = (size_t)R_REL * N_NODES * HD;
    if (i < nh) {
        segmax[i] = __int_as_float(0xFF800000);  // -inf
        segsum[i] = 0.0f;
    }
    if (i < nhd) agg[i] = 0.0f;
}

// ==================== GEMM stage A: f = x@Wn + bn, res = x@Wres + bres =========
// grid = (NBLK, R), block = 256 (8 waves). B fragments register-resident for the
// whole block; x tiles streamed via async global->LDS with double buffering.
__global__ __launch_bounds__(256) void k_gemm_fres(
    const float* __restrict__ x,     // (R,N,128)
    const float* __restrict__ Wn,    // (128,128)
    const float* __restrict__ bn,    // (128)
    const float* __restrict__ Wres,  // (128,128)
    const float* __restrict__ bres,  // (128)
    float* __restrict__ f_out,       // dst_after region (R,N,128)
    float* __restrict__ res_out)     // ws (R,N,128)
{
    __shared__ float xs[2][16 * 128];
    __shared__ float obuf[16 * 128];

    const int r    = blockIdx.y;
    const int tid  = threadIdx.x;
    const int lane = tid & 31;
    const int wave = tid >> 5;
    const int c0   = wave * 16;

    const int m   = lane & 15;            // A row
    const int kg  = (lane >> 4) << 1;     // K sub-offset: 0 / 2
    const int col = lane & 15;            // B/C column within tile
    const int rb  = (lane >> 4) << 3;     // C row base: 0 / 8

    // ---- preload B fragments (64 + 64 VGPRs), reused for all TILES tiles ----
    v2f bWn[32], bWrs[32];
#pragma unroll
    for (int kk = 0; kk < 32; ++kk) {
        int k = kk * 4 + kg;
        bWn[kk].x  = Wn  [(k    ) * HD + c0 + col];
        bWn[kk].y  = Wn  [(k + 1) * HD + c0 + col];
        bWrs[kk].x = Wres[(k    ) * HD + c0 + col];
        bWrs[kk].y = Wres[(k + 1) * HD + c0 + col];
    }
    const float biasN = bn[c0 + col];
    const float biasR = bres[c0 + col];

    const float* xr = x + (size_t)r * N_NODES * IN_DIM;
    const int tile0 = blockIdx.x * TILES;

    const uint32_t lb0 = lds_off32(&xs[0][0]) + tid * 32;
    const uint32_t lb1 = lds_off32(&xs[1][0]) + tid * 32;

    async_load_32B(lb0, xr, (uint32_t)tile0 * 8192u + tid * 32);

#pragma unroll 1
    for (int t = 0; t < TILES; ++t) {
        const int b = t & 1;
        if (t + 1 < TILES) {
            async_load_32B(b ? lb0 : lb1, xr,
                           (uint32_t)(tile0 + t + 1) * 8192u + tid * 32);
            WAIT_ASYNCCNT(0x2);
        } else {
            WAIT_ASYNCCNT(0x0);
        }
        __syncthreads();

        const float* xb = &xs[b][0];

        // ---- f = x @ Wn ----
        v8f c1 = {};
#pragma unroll
        for (int kk = 0; kk < 32; ++kk) {
            v2f a;
            a.x = xb[m * 128 + kk * 4 + kg];
            a.y = xb[m * 128 + kk * 4 + kg + 1];
            c1 = __builtin_amdgcn_wmma_f32_16x16x4_f32(false, a, false, bWn[kk],
                                                       (short)0, c1, false, false);
        }
#pragma unroll
        for (int v = 0; v < 8; ++v) obuf[(rb + v) * 128 + c0 + col] = c1[v] + biasN;
        __syncthreads();
        {
            float* fo = f_out + ((size_t)r * N_NODES + (size_t)(tile0 + t) * 16) * HD;
            for (int i = tid; i < 2048; i += 256) fo[i] = obuf[i];
        }

        // ---- res = x @ Wres ----
        v8f c3 = {};
#pragma unroll
        for (int kk = 0; kk < 32; ++kk) {
            v2f a;
            a.x = xb[m * 128 + kk * 4 + kg];
            a.y = xb[m * 128 + kk * 4 + kg + 1];
            c3 = __builtin_amdgcn_wmma_f32_16x16x4_f32(false, a, false, bWrs[kk],
                                                       (short)0, c3, false, false);
        }
        __syncthreads();   // f-store readers done with obuf
#pragma unroll
        for (int v = 0; v < 8; ++v) obuf[(rb + v) * 128 + c0 + col] = c3[v] + biasR;
        __syncthreads();
        {
            float* ro = res_out + ((size_t)r * N_NODES + (size_t)(tile0 + t) * 16) * HD;
            for (int i = tid; i < 2048; i += 256) ro[i] = obuf[i];
        }
        __syncthreads();   // end of iteration: protects xs[b] and obuf reuse
    }
}

// ==================== GEMM stage B: fs = f@Wr_r + br_r, attention logits =======
__global__ __launch_bounds__(256) void k_gemm_fs(
    const float* __restrict__ f_in,  // (R,N,128)  (dst_after region)
    const float* __restrict__ Wr,    // (R,128,128)
    const float* __restrict__ br,    // (R,128)
    const float* __restrict__ rwbuf, // (R,256) [h][0..15]=dst, [16..31]=src
    float* __restrict__ fs_out,      // ws (R,N,128)
    float* __restrict__ esrc,        // ws (R,N,H)
    float* __restrict__ edst)        // ws (R,N,H)
{
    __shared__ float fb[2][16 * 128];
    __shared__ float obuf[16 * 128];

    const int r    = blockIdx.y;
    const int tid  = threadIdx.x;
    const int lane = tid & 31;
    const int wave = tid >> 5;
    const int c0   = wave * 16;

    const int m   = lane & 15;
    const int kg  = (lane >> 4) << 1;
    const int col = lane & 15;
    const int rb  = (lane >> 4) << 3;

    const float* Wrr = Wr + (size_t)r * HD * HD;
    v2f bW[32];
#pragma unroll
    for (int kk = 0; kk < 32; ++kk) {
        int k = kk * 4 + kg;
        bW[kk].x = Wrr[(k    ) * HD + c0 + col];
        bW[kk].y = Wrr[(k + 1) * HD + c0 + col];
    }
    const float bias = br[r * HD + c0 + col];

    const float* fr = f_in + (size_t)r * N_NODES * HD;
    const int tile0 = blockIdx.x * TILES;

    const uint32_t lb0 = lds_off32(&fb[0][0]) + tid * 32;
    const uint32_t lb1 = lds_off32(&fb[1][0]) + tid * 32;

    async_load_32B(lb0, fr, (uint32_t)tile0 * 8192u + tid * 32);

#pragma unroll 1
    for (int t = 0; t < TILES; ++t) {
        const int b = t & 1;
        if (t + 1 < TILES) {
            async_load_32B(b ? lb0 : lb1, fr,
                           (uint32_t)(tile0 + t + 1) * 8192u + tid * 32);
            WAIT_ASYNCCNT(0x2);
        } else {
            WAIT_ASYNCCNT(0x0);
        }
        __syncthreads();

        const float* fbp = &fb[b][0];

        v8f c2 = {};
#pragma unroll
        for (int kk = 0; kk < 32; ++kk) {
            v2f a;
            a.x = fbp[m * 128 + kk * 4 + kg];
            a.y = fbp[m * 128 + kk * 4 + kg + 1];
            c2 = __builtin_amdgcn_wmma_f32_16x16x4_f32(false, a, false, bW[kk],
                                                       (short)0, c2, false, false);
        }
        __syncthreads();   // previous iteration's obuf readers done
#pragma unroll
        for (int v = 0; v < 8; ++v) obuf[(rb + v) * 128 + c0 + col] = c2[v] + bias;
        __syncthreads();
        {
            float* fso = fs_out + ((size_t)r * N_NODES + (size_t)(tile0 + t) * 16) * HD;
            for (int i = tid; i < 2048; i += 256) fso[i] = obuf[i];
        }
        if (tid < 128) {   // e_src from fs (obuf), e_dst from f (fbp)
            int row = tid >> 3, h = tid & 7;
            const float* rw = rwbuf + r * 256 + h * 32;
            float ss = 0.0f, ds = 0.0f;
#pragma unroll
            for (int d = 0; d < D; ++d) {
                ss += obuf[row * 128 + h * D + d] * rw[D + d];
                ds += fbp [row * 128 + h * D + d] * rw[d];
            }
            size_t base = ((size_t)r * N_NODES + (size_t)(tile0 + t) * 16 + row) * H + h;
            esrc[base] = ss;
            edst[base] = ds;
        }
        __syncthreads();   // end of iteration
    }
}

// ---------------- edge pass 1: segment max ----------------
__global__ __launch_bounds__(256) void k_edge_max(
    const int* __restrict__ src_idx, const int* __restrict__ dst_idx,
    const float* __restrict__ esrc, const float* __restrict__ edst,
    float* __restrict__ segmax)
{
    int idx = blockIdx.x * blockDim.x + threadIdx.x;
    if (idx >= R_REL * E_EDGES) return;
    int r = idx / E_EDGES, e = idx - r * E_EDGES;
    int s = src_idx[r * E_EDGES + e], d = dst_idx[r * E_EDGES + e];
    const float* es = esrc + ((size_t)r * N_NODES + s) * H;
    const float* ed = edst + ((size_t)r * N_NODES + d) * H;
    float* sm = segmax + ((size_t)r * N_NODES + d) * H;
#pragma unroll
    for (int h = 0; h < H; ++h) atomicMaxF(&sm[h], leaky(es[h] + ed[h]));
}

// ---------------- edge pass 2: exp + segment sum ----------------
__global__ __launch_bounds__(256) void k_edge_exp(
    const int* __restrict__ src_idx, const int* __restrict__ dst_idx,
    const float* __restrict__ esrc, const float* __restrict__ edst,
    const float* __restrict__ segmax, float* __restrict__ aun,
    float* __restrict__ segsum)
{
    int idx = blockIdx.x * blockDim.x + threadIdx.x;
    if (idx >= R_REL * E_EDGES) return;
    int r = idx / E_EDGES, e = idx - r * E_EDGES;
    int s = src_idx[r * E_EDGES + e], d = dst_idx[r * E_EDGES + e];
    const float* es = esrc + ((size_t)r * N_NODES + s) * H;
    const float* ed = edst + ((size_t)r * N_NODES + d) * H;
    const float* sm = segmax + ((size_t)r * N_NODES + d) * H;
    float* ss = segsum + ((size_t)r * N_NODES + d) * H;
    float* au = aun + ((size_t)r * E_EDGES + e) * H;
#pragma unroll
    for (int h = 0; h < H; ++h) {
        float a = __expf(leaky(es[h] + ed[h]) - sm[h]);
        au[h] = a;
        atomicAddF(&ss[h], a);
    }
}

// ---------------- edge pass 3: normalize + gather fs[src], scatter-add agg[dst]
__global__ __launch_bounds__(256) void k_edge_agg(
    const int* __restrict__ src_idx, const int* __restrict__ dst_idx,
    const float* __restrict__ fs, const float* __restrict__ aun,
    const float* __restrict__ segsum, float* __restrict__ agg)
{
    size_t idx = (size_t)blockIdx.x * blockDim.x + threadIdx.x;
    if (idx >= (size_t)R_REL * E_EDGES * HD) return;
    int cidx = (int)(idx & 127);
    size_t t = idx >> 7;
    int e = (int)(t % E_EDGES);
    int r = (int)(t / E_EDGES);
    int s = src_idx[r * E_EDGES + e], d = dst_idx[r * E_EDGES + e];
    int h = cidx >> 4;
    float a = aun[((size_t)r * E_EDGES + e) * H + h];
    float z = segsum[((size_t)r * N_NODES + d) * H + h];
    a = a / (z + 1e-16f);
    float v = fs[((size_t)r * N_NODES + s) * HD + cidx] * a;
    atomicAddF(&agg[((size_t)r * N_NODES + d) * HD + cidx], v);
}

// ---------------- gated residual blend (in-place on out region) ----------------
__global__ __launch_bounds__(256) void k_blend(
    float* __restrict__ out, const float* __restrict__ res,
    const float* __restrict__ res_w)
{
    size_t i = (size_t)blockIdx.x * blockDim.x + threadIdx.x;
    if (i >= (size_t)R_REL * N_NODES * HD) return;
    float alpha = 1.0f / (1.0f + __expf(-res_w[0]));
    float a = out[i];
    a = a > 0.0f ? a : 0.0f;                // relu(agg)
    out[i] = a * alpha + res[i] * (1.0f - alpha);
}

// ---------------- cross-relation attention (in-place, thread owns (n,h)) ------
__global__ __launch_bounds__(256) void k_cross(
    float* __restrict__ out, const float* __restrict__ cross_w)
{
    int idx = blockIdx.x * blockDim.x + threadIdx.x;
    if (idx >= N_NODES * H) return;
    int n = idx >> 3, h = idx & 7;

    float feats[R_REL][D];
#pragma unroll
    for (int r = 0; r < R_REL; ++r)
#pragma unroll
        for (int d = 0; d < D; ++d)
            feats[r][d] = out[((size_t)r * N_NODES + n) * HD + h * D + d];

#pragma unroll
    for (int er = 0; er < R_REL; ++er) {
        float sc[R_REL];
#pragma unroll
        for (int r = 0; r < R_REL; ++r) {
            float s = 0.0f;
#pragma unroll
            for (int d = 0; d < D; ++d)
                s += feats[r][d] * cross_w[(er * H + h) * D + d];
            sc[r] = leaky(s);
        }
        float mx = fmaxf(sc[0], fmaxf(sc[1], sc[2]));
        float w[R_REL], z = 0.0f;
#pragma unroll
        for (int r = 0; r < R_REL; ++r) { w[r] = __expf(sc[r] - mx); z += w[r]; }
        float inv = 1.0f / z;
#pragma unroll
        for (int d = 0; d < D; ++d) {
            float acc = 0.0f;
#pragma unroll
            for (int r = 0; r < R_REL; ++r) acc += w[r] * inv * feats[r][d];
            out[((size_t)er * N_NODES + n) * HD + h * D + d] = acc;
        }
    }
}

// ---------------- host-side launcher ----------------
extern "C" void kernel_launch(void* const* d_in, const int* in_sizes, int n_in,
                              void* d_out, int out_size, void* d_ws, size_t ws_size,
                              hipStream_t stream) {
    const float* x          = (const float*)d_in[0];
    const float* rel_emb    = (const float*)d_in[1];
    const int*   src_idx    = (const int*)  d_in[2];
    const int*   dst_idx    = (const int*)  d_in[3];
    const float* Wn         = (const float*)d_in[4];
    const float* bn         = (const float*)d_in[5];
    const float* Wr         = (const float*)d_in[6];
    const float* br         = (const float*)d_in[7];
    const float* rel_attn_w = (const float*)d_in[8];
    const float* Wp         = (const float*)d_in[9];
    const float* bp         = (const float*)d_in[10];
    const float* Wres       = (const float*)d_in[11];
    const float* bres       = (const float*)d_in[12];
    const float* res_w      = (const float*)d_in[13];
    const float* cross_w    = (const float*)d_in[14];

    const size_t NHD = (size_t)R_REL * N_NODES * HD;   // 38,400,000
    const size_t NH  = (size_t)R_REL * N_NODES * H;    //  2,400,000
    const size_t EH  = (size_t)R_REL * E_EDGES * H;    //  9,600,000

    float* out     = (float*)d_out;                     // (R,N,HD): agg -> final
    float* rel_out = out + NHD;                         // (R,64)
    float* f_out   = rel_out + R_REL * H * REL_HID;     // dst_after (R,N,HD)

    float* ws = (float*)d_ws;
    size_t o = 0;
    float* fs     = ws + o; o += NHD;
    float* res    = ws + o; o += NHD;
    float* esrc   = ws + o; o += NH;
    float* edst   = ws + o; o += NH;
    float* segmax = ws + o; o += NH;
    float* segsum = ws + o; o += NH;
    float* aun    = ws + o; o += EH;
    float* rwbuf  = ws + o; o += R_REL * 2 * HD;

    // 1. tiny precompute (rw, rel_out)
    k_small<<<1, 256, 0, stream>>>(rel_emb, rel_attn_w, Wp, bp, rwbuf, rel_out);

    // 2. init segmax/segsum/agg
    {
        int blocks = (int)((NHD + 255) / 256);
        k_init<<<blocks, 256, 0, stream>>>(segmax, segsum, out);
    }

    // 3a. f and res (WMMA f32, register-resident B, async double-buffered tiles)
    k_gemm_fres<<<dim3(NBLK, R_REL), 256, 0, stream>>>(
        x, Wn, bn, Wres, bres, f_out, res);

    // 3b. fs + attention logits
    k_gemm_fs<<<dim3(NBLK, R_REL), 256, 0, stream>>>(
        f_out, Wr, br, rwbuf, fs, esrc, edst);

    // 4-6. edge softmax + aggregation
    {
        int eb = (R_REL * E_EDGES + 255) / 256;
        k_edge_max<<<eb, 256, 0, stream>>>(src_idx, dst_idx, esrc, edst, segmax);
        k_edge_exp<<<eb, 256, 0, stream>>>(src_idx, dst_idx, esrc, edst, segmax,
                                           aun, segsum);
        size_t total = (size_t)R_REL * E_EDGES * HD;
        int ab = (int)((total + 255) / 256);
        k_edge_agg<<<ab, 256, 0, stream>>>(src_idx, dst_idx, fs, aun, segsum, out);
    }

    // 7. gated residual blend
    {
        int blocks = (int)((NHD + 255) / 256);
        k_blend<<<blocks, 256, 0, stream>>>(out, res, res_w);
    }

    // 8. cross-relation attention (in-place)
    {
        int blocks = (N_NODES * H + 255) / 256;
        k_cross<<<blocks, 256, 0, stream>>>(out, cross_w);
    }
}